// GNNGlobalAttention_807453851813
// MI455X (gfx1250) — compile-verified
//
#include <hip/hip_runtime.h>

// ---- problem constants (match reference) ----
#define FEAT   128
#define NCLS   16
#define NGRAPH 128

typedef __attribute__((ext_vector_type(2))) float v2f;
typedef __attribute__((ext_vector_type(4))) float v4f;
typedef __attribute__((ext_vector_type(8))) float v8f;

// ---------- small utility kernels ----------
__global__ void k_fill(float* __restrict__ p, float v, int n) {
    int i = blockIdx.x * blockDim.x + threadIdx.x;
    if (i < n) p[i] = v;
}

__global__ void k_degree(const int* __restrict__ dst, float* __restrict__ deg, int E) {
    int i = blockIdx.x * blockDim.x + threadIdx.x;
    if (i < E) atomicAdd(&deg[dst[i]], 1.0f);
}

__global__ void k_rsqrt_inplace(float* __restrict__ d, int n) {
    int i = blockIdx.x * blockDim.x + threadIdx.x;
    if (i < n) d[i] = rsqrtf(d[i]);   // deg >= 1 always (self-loop)
}

// ---------- fp32 WMMA GEMM:  C[M x 128] = A[M x 128] * B[128 x 128] ----------
// One wave computes a full 16x128 row-block: 8 accumulators, so each A
// fragment (loaded once per K-step) feeds 8 independent
// v_wmma_f32_16x16x4_f32 ops. Layouts per CDNA5 ISA 7.12.2:
//   A 16x4 fp32 : lanes 0-15 hold K=0,1 of row m; lanes 16-31 hold K=2,3 (v2f)
//   B 4x16 fp32 : VGPR0 = K=0 (lanes 0-15) / K=2 (lanes 16-31); VGPR1 = K=1 / K=3
//   C/D 16x16   : VGPR v = row v (lanes 0-15) / row v+8 (lanes 16-31)
template <int FUSE_BIAS_RELU>
__global__ void k_gemm128_wmma(const float* __restrict__ A,
                               const float* __restrict__ B,
                               const float* __restrict__ bias,
                               float* __restrict__ C,
                               int M) {
    const int wave  = threadIdx.x >> 5;
    const int lane  = threadIdx.x & 31;
    const int half  = lane >> 4;      // 0 for lanes 0-15, 1 for lanes 16-31
    const int l15   = lane & 15;
    const int tileM = (blockIdx.x * (blockDim.x >> 5) + wave) * 16;
    if (tileM >= M) return;           // whole-wave exit before any WMMA

    int m = tileM + l15;
    if (m >= M) m = M - 1;            // clamp loads; keep EXEC all-ones for WMMA
    const float* arow = A + (size_t)m * FEAT;

    v8f acc[8];
    #pragma unroll
    for (int j = 0; j < 8; ++j) acc[j] = (v8f){};

    for (int k0 = 0; k0 < FEAT; k0 += 4) {
        const int ka = k0 + 2 * half;                 // even -> 8B-aligned
        const v2f a = *(const v2f*)(arow + ka);       // A[m][ka], A[m][ka+1]
        const float* brow0 = B + (size_t)ka * FEAT + l15;
        const float* brow1 = brow0 + FEAT;
        #pragma unroll
        for (int j = 0; j < 8; ++j) {                 // 8 N-tiles reuse one A frag
            v2f b;
            b.x = brow0[j * 16];
            b.y = brow1[j * 16];
            acc[j] = __builtin_amdgcn_wmma_f32_16x16x4_f32(
                         false, a, false, b, (short)0, acc[j], false, false);
        }
    }

    #pragma unroll
    for (int j = 0; j < 8; ++j) {
        const int n = j * 16 + l15;
        const float bval = FUSE_BIAS_RELU ? bias[n] : 0.0f;
        #pragma unroll
        for (int v = 0; v < 8; ++v) {
            const int row = tileM + v + half * 8;
            if (row < M) {
                float x = acc[j][v];
                if (FUSE_BIAS_RELU) x = fmaxf(x + bval, 0.0f);
                C[(size_t)row * FEAT + n] = x;
            }
        }
    }
}

// ---------- edge gather-scale-scatter: one wave per edge ----------
__global__ void k_edge_agg(const float* __restrict__ tmp,
                           const int* __restrict__ src,
                           const int* __restrict__ dst,
                           const float* __restrict__ dis,
                           float* __restrict__ agg, int E) {
    const int e = blockIdx.x * (blockDim.x >> 5) + (threadIdx.x >> 5);
    if (e >= E) return;
    const int lane = threadIdx.x & 31;
    const int s = src[e], d = dst[e];
    const float norm = dis[s] * dis[d];
    const v4f val = *(const v4f*)(tmp + (size_t)s * FEAT + lane * 4);
    float* od = agg + (size_t)d * FEAT + lane * 4;
    atomicAdd(&od[0], val.x * norm);
    atomicAdd(&od[1], val.y * norm);
    atomicAdd(&od[2], val.z * norm);
    atomicAdd(&od[3], val.w * norm);
}

// ---------- self-loop + bias + relu fused finalize ----------
__global__ void k_finalize(float* __restrict__ agg, const float* __restrict__ tmp,
                           const float* __restrict__ dis, const float* __restrict__ bias,
                           int total /* N*FEAT */) {
    int i = blockIdx.x * blockDim.x + threadIdx.x;
    if (i >= total) return;
    const int node = i >> 7;
    const int f    = i & (FEAT - 1);
    const float sn = dis[node] * dis[node];
    agg[i] = fmaxf(agg[i] + tmp[i] * sn + bias[f], 0.0f);
}

// ---------- order-preserving float<->uint for atomic max ----------
__device__ __forceinline__ unsigned ordf(float f) {
    unsigned u = __float_as_uint(f);
    return (u & 0x80000000u) ? ~u : (u | 0x80000000u);
}
__device__ __forceinline__ float unordf(unsigned u) {
    unsigned v = (u & 0x80000000u) ? (u & 0x7FFFFFFFu) : ~u;
    return __uint_as_float(v);
}

// ---------- gate scalar (one wave/node): g[i] = dot(gateH[i], GW2) + Gb2 ----------
__global__ void k_gate(const float* __restrict__ gateH, const float* __restrict__ GW2,
                       const float* __restrict__ Gb2, const int* __restrict__ batch,
                       float* __restrict__ g, unsigned* __restrict__ gmax_u, int N) {
    const int node = blockIdx.x * (blockDim.x >> 5) + (threadIdx.x >> 5);
    if (node >= N) return;
    const int lane = threadIdx.x & 31;
    const v4f hv = *(const v4f*)(gateH + (size_t)node * FEAT + lane * 4);
    const v4f wv = *(const v4f*)(GW2 + lane * 4);
    float s = hv.x * wv.x + hv.y * wv.y + hv.z * wv.z + hv.w * wv.w;
    #pragma unroll
    for (int off = 16; off > 0; off >>= 1) s += __shfl_down(s, off, 32);
    if (lane == 0) {
        s += Gb2[0];
        g[node] = s;
        atomicMax(&gmax_u[batch[node]], ordf(s));
    }
}

// ---------- e = exp(g - gmax[batch]); denom segment sum ----------
__global__ void k_softmax_e(float* __restrict__ g, const unsigned* __restrict__ gmax_u,
                            const int* __restrict__ batch, float* __restrict__ denom, int N) {
    int i = blockIdx.x * blockDim.x + threadIdx.x;
    if (i >= N) return;
    const int b = batch[i];
    const float e = expf(g[i] - unordf(gmax_u[b]));
    g[i] = e;
    atomicAdd(&denom[b], e);
}

// ---------- pooled[b] += alpha * h[i]  (one wave per node) ----------
__global__ void k_pool(const float* __restrict__ h, const float* __restrict__ e,
                       const float* __restrict__ denom, const int* __restrict__ batch,
                       float* __restrict__ pooled, int N) {
    const int node = blockIdx.x * (blockDim.x >> 5) + (threadIdx.x >> 5);
    if (node >= N) return;
    const int lane = threadIdx.x & 31;
    const int b = batch[node];
    const float alpha = e[node] / denom[b];
    const v4f hv = *(const v4f*)(h + (size_t)node * FEAT + lane * 4);
    float* pd = pooled + (size_t)b * FEAT + lane * 4;
    atomicAdd(&pd[0], hv.x * alpha);
    atomicAdd(&pd[1], hv.y * alpha);
    atomicAdd(&pd[2], hv.z * alpha);
    atomicAdd(&pd[3], hv.w * alpha);
}

// ---------- out[g][c] = pooled[g] . LW[:,c] + Lb[c] ----------
__global__ void k_out(const float* __restrict__ pooled, const float* __restrict__ LW,
                      const float* __restrict__ Lb, float* __restrict__ out) {
    int t = blockIdx.x * blockDim.x + threadIdx.x;
    if (t >= NGRAPH * NCLS) return;
    const int g = t >> 4, c = t & (NCLS - 1);
    float s = Lb[c];
    #pragma unroll 8
    for (int k = 0; k < FEAT; ++k) s += pooled[(size_t)g * FEAT + k] * LW[k * NCLS + c];
    out[t] = s;
}

static inline int cdiv(int a, int b) { return (a + b - 1) / b; }

extern "C" void kernel_launch(void* const* d_in, const int* in_sizes, int n_in,
                              void* d_out, int out_size, void* d_ws, size_t ws_size,
                              hipStream_t stream) {
    const float* x    = (const float*)d_in[0];
    const int*   ei   = (const int*)  d_in[1];
    const int*   bat  = (const int*)  d_in[2];
    const float* W1   = (const float*)d_in[3];
    const float* b1   = (const float*)d_in[4];
    const float* W2   = (const float*)d_in[5];
    const float* b2   = (const float*)d_in[6];
    const float* GW1  = (const float*)d_in[7];
    const float* Gb1  = (const float*)d_in[8];
    const float* GW2  = (const float*)d_in[9];
    const float* Gb2  = (const float*)d_in[10];
    const float* LW   = (const float*)d_in[11];
    const float* Lb   = (const float*)d_in[12];
    float* out = (float*)d_out;

    const int N = in_sizes[0] / FEAT;      // 100000
    const int E = in_sizes[1] / 2;         // 1600000
    const int* src = ei;
    const int* dst = ei + E;

    // ---- workspace carve-up ----
    float*    bufA   = (float*)d_ws;                      // N*FEAT  (lin result / gate hidden)
    float*    bufB   = bufA + (size_t)N * FEAT;           // N*FEAT  (aggregated / h)
    float*    dis    = bufB + (size_t)N * FEAT;           // N
    float*    gbuf   = dis + N;                           // N  (gate scalar, then e)
    float*    denom  = gbuf + N;                          // NGRAPH
    unsigned* gmaxu  = (unsigned*)(denom + NGRAPH);       // NGRAPH
    float*    pooled = (float*)(gmaxu + NGRAPH);          // NGRAPH*FEAT

    const size_t featBytes = (size_t)N * FEAT * sizeof(float);

    // ---- zero/init small state ----
    hipMemsetAsync(denom, 0, NGRAPH * sizeof(float), stream);
    hipMemsetAsync(gmaxu, 0, NGRAPH * sizeof(unsigned), stream);   // below ord() of any real
    hipMemsetAsync(pooled, 0, (size_t)NGRAPH * FEAT * sizeof(float), stream);

    // ---- symmetric normalization: dis = rsqrt(indeg + 1) ----
    k_fill<<<cdiv(N, 256), 256, 0, stream>>>(dis, 1.0f, N);
    k_degree<<<cdiv(E, 256), 256, 0, stream>>>(dst, dis, E);
    k_rsqrt_inplace<<<cdiv(N, 256), 256, 0, stream>>>(dis, N);

    const int gemmBlocks = cdiv(cdiv(N, 16), 8);   // 8 waves/block, 16 rows/wave
    const int edgeBlocks = cdiv(E, 8);             // 8 waves (edges) per 256-thread block
    const int nodeBlocks = cdiv(N, 8);

    // ---- GCN layer 1 ----
    k_gemm128_wmma<0><<<gemmBlocks, 256, 0, stream>>>(x, W1, nullptr, bufA, N);
    hipMemsetAsync(bufB, 0, featBytes, stream);
    k_edge_agg<<<edgeBlocks, 256, 0, stream>>>(bufA, src, dst, dis, bufB, E);
    k_finalize<<<cdiv(N * FEAT, 256), 256, 0, stream>>>(bufB, bufA, dis, b1, N * FEAT);

    // ---- GCN layer 2 (bufB = h1 is fully consumed by the GEMM before re-zeroing) ----
    k_gemm128_wmma<0><<<gemmBlocks, 256, 0, stream>>>(bufB, W2, nullptr, bufA, N);
    hipMemsetAsync(bufB, 0, featBytes, stream);
    k_edge_agg<<<edgeBlocks, 256, 0, stream>>>(bufA, src, dst, dis, bufB, E);
    k_finalize<<<cdiv(N * FEAT, 256), 256, 0, stream>>>(bufB, bufA, dis, b2, N * FEAT);
    // bufB now holds h2

    // ---- gate MLP: bufA = relu(h2 @ GW1 + Gb1) ----
    k_gemm128_wmma<1><<<gemmBlocks, 256, 0, stream>>>(bufB, GW1, Gb1, bufA, N);
    k_gate<<<nodeBlocks, 256, 0, stream>>>(bufA, GW2, Gb2, bat, gbuf, gmaxu, N);

    // ---- per-graph softmax + attention pooling ----
    k_softmax_e<<<cdiv(N, 256), 256, 0, stream>>>(gbuf, gmaxu, bat, denom, N);
    k_pool<<<nodeBlocks, 256, 0, stream>>>(bufB, gbuf, denom, bat, pooled, N);

    // ---- classifier head ----
    k_out<<<cdiv(NGRAPH * NCLS, 256), 256, 0, stream>>>(pooled, LW, Lb, out);
}